// DynamicVFE_26834955665466
// MI455X (gfx1250) — compile-verified
//
#include <hip/hip_runtime.h>
#include <hip/hip_bf16.h>

typedef __attribute__((ext_vector_type(2))) float v2f;
typedef __attribute__((ext_vector_type(8))) float v8f;

// ---------------- voxel xyz sums + counts (segment_sum) ----------------
__global__ void k_vox_sum(const float* __restrict__ feat, const int* __restrict__ inv,
                          float* __restrict__ vox, int n) {
    int i = blockIdx.x * 256 + threadIdx.x;
    if (i >= n) return;
    int v = inv[i];
    atomicAdd(&vox[(size_t)v * 4 + 0], feat[(size_t)i * 4 + 0]);
    atomicAdd(&vox[(size_t)v * 4 + 1], feat[(size_t)i * 4 + 1]);
    atomicAdd(&vox[(size_t)v * 4 + 2], feat[(size_t)i * 4 + 2]);
    atomicAdd(&vox[(size_t)v * 4 + 3], 1.0f);
}

// ---------------- GEMM0: x10[N,10] @ W0^T[10,64] -> h0pre[N,64] ----------------
// block = 256 threads = 4 rows x 64 cols; x10 staged in LDS per row.
__global__ void k_gemm0(const float* __restrict__ feat, const int* __restrict__ inv,
                        const int* __restrict__ coors, const float* __restrict__ vox,
                        const float* __restrict__ W0, float* __restrict__ h0, int n) {
    __shared__ float w[640];
    __shared__ float sx[4][10];
    for (int t = threadIdx.x; t < 640; t += 256) w[t] = W0[t];
    int r = threadIdx.x >> 6;   // 0..3
    int c = threadIdx.x & 63;   // 0..63
    int p = blockIdx.x * 4 + r;
    if (c == 0 && p < n) {
        float x = feat[(size_t)p * 4 + 0];
        float y = feat[(size_t)p * 4 + 1];
        float z = feat[(size_t)p * 4 + 2];
        int v = inv[p];
        float cnt = vox[(size_t)v * 4 + 3];
        float rc = 1.0f / cnt;
        float mx = vox[(size_t)v * 4 + 0] * rc;
        float my = vox[(size_t)v * 4 + 1] * rc;
        float mz = vox[(size_t)v * 4 + 2] * rc;
        // coors order (b, z, y, x); voxel sizes 0.2,0.2,4.0; mins 0,-40,-3
        float cx = (float)coors[(size_t)v * 4 + 3] * 0.2f + 0.1f + 0.0f;
        float cy = (float)coors[(size_t)v * 4 + 2] * 0.2f + 0.1f - 40.0f;
        float cz = (float)coors[(size_t)v * 4 + 1] * 4.0f + 2.0f - 3.0f;
        sx[r][0] = x;      sx[r][1] = y;      sx[r][2] = z;
        sx[r][3] = x - mx; sx[r][4] = y - my; sx[r][5] = z - mz;
        sx[r][6] = x - cx; sx[r][7] = y - cy; sx[r][8] = z - cz;
        sx[r][9] = feat[(size_t)p * 4 + 3];
    }
    __syncthreads();
    if (p < n) {
        float acc = 0.0f;
#pragma unroll
        for (int k = 0; k < 10; ++k) acc += sx[r][k] * w[c * 10 + k];
        h0[(size_t)p * 64 + c] = acc;
    }
}

// ---------------- per-column sum / sumsq over [n,64] ----------------
__global__ void k_stats(const float* __restrict__ buf, float* __restrict__ sum,
                        float* __restrict__ sumsq, int n) {
    __shared__ float ss[64], sq[64];
    if (threadIdx.x < 64) { ss[threadIdx.x] = 0.0f; sq[threadIdx.x] = 0.0f; }
    __syncthreads();
    int col = threadIdx.x & 63;
    int r = blockIdx.x * 4 + (threadIdx.x >> 6);
    int rstep = gridDim.x * 4;
    float a = 0.0f, b = 0.0f;
    for (; r < n; r += rstep) {
        float x = buf[(size_t)r * 64 + col];
        a += x;
        b += x * x;
    }
    atomicAdd(&ss[col], a);
    atomicAdd(&sq[col], b);
    __syncthreads();
    if (threadIdx.x < 64) {
        atomicAdd(&sum[threadIdx.x], ss[threadIdx.x]);
        atomicAdd(&sumsq[threadIdx.x], sq[threadIdx.x]);
    }
}

// ---------------- BN(train) + ReLU + segment_max (int-bit atomicMax, vals >= 0) ----------------
__global__ void k_bn_max(float* __restrict__ buf, const float* __restrict__ sum,
                         const float* __restrict__ sumsq, const float* __restrict__ gamma,
                         const float* __restrict__ beta, const int* __restrict__ inv,
                         int* __restrict__ dst, int n, float invN) {
    __shared__ float sc[64], sh[64];
    if (threadIdx.x < 64) {
        float mu  = sum[threadIdx.x] * invN;
        float var = sumsq[threadIdx.x] * invN - mu * mu;
        float rs  = rsqrtf(var + 1e-3f);
        sc[threadIdx.x] = gamma[threadIdx.x] * rs;
        sh[threadIdx.x] = beta[threadIdx.x] - gamma[threadIdx.x] * rs * mu;
    }
    __syncthreads();
    size_t idx = (size_t)blockIdx.x * 256 + threadIdx.x;
    if (idx >= (size_t)n * 64) return;
    int col = (int)(idx & 63);
    int row = (int)(idx >> 6);
    float x = fmaxf(buf[idx] * sc[col] + sh[col], 0.0f);
    buf[idx] = x;
    atomicMax(&dst[(size_t)inv[row] * 64 + col], __float_as_int(x));
}

// ---------------- GEMM1 via f32 WMMA: x1[N,128] @ W1^T[128,64] -> h1pre[N,64] ----
// 128 threads = 4 waves; each wave owns a 16-point tile, K=128 in 32 chunks of 4,
// 4 output tiles of 16 cols -> 128 v_wmma_f32_16x16x4_f32 per wave.
__global__ void __launch_bounds__(128) k_gemm1(const float* __restrict__ h0,
                                               const float* __restrict__ v0,
                                               const int* __restrict__ inv,
                                               const float* __restrict__ W1,
                                               float* __restrict__ h1, int n) {
    __shared__ float w1[64 * 128];     // 32 KB: W1[ncol][k], row-major as given
    __shared__ float xs[4][16][128];   // 32 KB: x1 tile per wave
    for (int t = threadIdx.x; t < 64 * 128; t += 128) w1[t] = W1[t];
    int wave = threadIdx.x >> 5;
    int lane = threadIdx.x & 31;
    int p0 = (blockIdx.x * 4 + wave) * 16;

    for (int r = 0; r < 16; ++r) {
        int p = p0 + r;
        if (p < n) {
            int v = inv[p];
            xs[wave][r][lane]      = h0[(size_t)p * 64 + lane];
            xs[wave][r][lane + 32] = h0[(size_t)p * 64 + lane + 32];
            xs[wave][r][64 + lane] = v0[(size_t)v * 64 + lane];
            xs[wave][r][96 + lane] = v0[(size_t)v * 64 + lane + 32];
        } else {
            xs[wave][r][lane] = 0.0f;
            xs[wave][r][lane + 32] = 0.0f;
            xs[wave][r][64 + lane] = 0.0f;
            xs[wave][r][96 + lane] = 0.0f;
        }
    }
    __syncthreads();

    int m = lane & 15;                 // row (A) / col (B) within tile
    int koff = (lane >> 4) << 1;       // 0 or 2: K offset per half-wave
    v8f c0 = {}, c1 = {}, c2 = {}, c3 = {};
    for (int ch = 0; ch < 32; ++ch) {
        int k0 = ch * 4 + koff;
        v2f a  = *(const v2f*)&xs[wave][m][k0];
        v2f b0 = *(const v2f*)&w1[(0  + m) * 128 + k0];
        v2f b1 = *(const v2f*)&w1[(16 + m) * 128 + k0];
        v2f b2 = *(const v2f*)&w1[(32 + m) * 128 + k0];
        v2f b3 = *(const v2f*)&w1[(48 + m) * 128 + k0];
        c0 = __builtin_amdgcn_wmma_f32_16x16x4_f32(false, a, false, b0, (short)0, c0, false, false);
        c1 = __builtin_amdgcn_wmma_f32_16x16x4_f32(false, a, false, b1, (short)0, c1, false, false);
        c2 = __builtin_amdgcn_wmma_f32_16x16x4_f32(false, a, false, b2, (short)0, c2, false, false);
        c3 = __builtin_amdgcn_wmma_f32_16x16x4_f32(false, a, false, b3, (short)0, c3, false, false);
    }

    int hi8 = (lane >> 4) << 3;        // C/D: lanes 16-31 hold M = j + 8
#pragma unroll
    for (int j = 0; j < 8; ++j) {
        int row = p0 + j + hi8;
        if (row < n) {
            size_t base = (size_t)row * 64 + m;
            h1[base + 0]  = c0[j];
            h1[base + 16] = c1[j];
            h1[base + 32] = c2[j];
            h1[base + 48] = c3[j];
        }
    }
}

// ---------------- coors passthrough ----------------
__global__ void k_copy_i32(const int* __restrict__ src, int* __restrict__ dst, int n) {
    int i = blockIdx.x * 256 + threadIdx.x;
    if (i < n) dst[i] = src[i];
}

extern "C" void kernel_launch(void* const* d_in, const int* in_sizes, int n_in,
                              void* d_out, int out_size, void* d_ws, size_t ws_size,
                              hipStream_t stream) {
    const float* feat = (const float*)d_in[0];
    const float* W0   = (const float*)d_in[1];
    const float* g0   = (const float*)d_in[2];
    const float* b0   = (const float*)d_in[3];
    const float* W1   = (const float*)d_in[4];
    const float* g1   = (const float*)d_in[5];
    const float* b1   = (const float*)d_in[6];
    const int*   inv  = (const int*)d_in[7];
    const int*   coor = (const int*)d_in[8];

    int n = in_sizes[0] / 4;   // points
    int v = in_sizes[8] / 4;   // voxels

    // workspace layout (floats)
    float* ws    = (float*)d_ws;
    float* vox   = ws;                          // V*4   (sumx,sumy,sumz,cnt)
    float* stats = vox + (size_t)v * 4;         // 256   (sum0,sq0,sum1,sq1)
    float* h0    = stats + 256;                 // N*64
    float* v0    = h0 + (size_t)n * 64;         // V*64
    float* h1    = v0 + (size_t)v * 64;         // N*64

    float* outf = (float*)d_out;                // [V,64] voxel feats
    int*   outi = (int*)d_out;                  // coors at offset V*64

    // zero accumulators (harness poisons ws/out once; we re-init every call)
    hipMemsetAsync(vox, 0, sizeof(float) * ((size_t)v * 4 + 256), stream);
    hipMemsetAsync(v0, 0, sizeof(float) * (size_t)v * 64, stream);
    hipMemsetAsync(outf, 0, sizeof(float) * (size_t)v * 64, stream);

    float invN = 1.0f / (float)n;

    k_vox_sum<<<(n + 255) / 256, 256, 0, stream>>>(feat, inv, vox, n);
    k_gemm0<<<(n + 3) / 4, 256, 0, stream>>>(feat, inv, coor, vox, W0, h0, n);
    k_stats<<<512, 256, 0, stream>>>(h0, stats + 0, stats + 64, n);
    k_bn_max<<<(int)(((size_t)n * 64 + 255) / 256), 256, 0, stream>>>(
        h0, stats + 0, stats + 64, g0, b0, inv, (int*)v0, n, invN);
    k_gemm1<<<(n + 63) / 64, 128, 0, stream>>>(h0, v0, inv, W1, h1, n);
    k_stats<<<512, 256, 0, stream>>>(h1, stats + 128, stats + 192, n);
    k_bn_max<<<(int)(((size_t)n * 64 + 255) / 256), 256, 0, stream>>>(
        h1, stats + 128, stats + 192, g1, b1, inv, outi, n, invN);
    k_copy_i32<<<(v * 4 + 255) / 256, 256, 0, stream>>>(coor, outi + (size_t)v * 64, v * 4);
}